// HGNNPConvL_56410100466017
// MI455X (gfx1250) — compile-verified
//
#include <hip/hip_runtime.h>
#include <cstdint>

typedef __attribute__((ext_vector_type(2))) float v2f;
typedef __attribute__((ext_vector_type(8))) float v8f;
typedef __attribute__((ext_vector_type(4))) float fvec4;

#define DCH 512          // D_IN == D_OUT == 512
#define BN_EPS 1e-5f
#define BS_STRIDE 34     // transposed B tile row stride (floats); 136B = 8B-aligned

// ---------------------------------------------------------------------------
// GEMM: Y[N,512] = X[N,512] @ W[512,512] + bias, fp32 via V_WMMA_F32_16X16X4_F32
// Block tile 64(M) x 128(N), 256 threads = 8 waves arranged 4(M) x 2(N).
// Each wave computes 16x64 (four 16x16 WMMA accumulators), K staged via LDS.
// B tile kept TRANSPOSED in LDS so both A and B fragments are single
// contiguous 8-byte LDS loads (ds_load_b64), no VGPR repacking before WMMA.
// ---------------------------------------------------------------------------
__global__ __launch_bounds__(256) void gemm_wmma_f32(
    const float* __restrict__ X, const float* __restrict__ W,
    const float* __restrict__ bias, float* __restrict__ Y, int Nrows) {
  __shared__ float As[64 * 32];            // [m][k], stride 32
  __shared__ float Bs[128 * BS_STRIDE];    // [n][k], stride 34 (transposed)

  const int tid  = threadIdx.x;
  const int lane = tid & 31;
  const int wave = tid >> 5;
  const int wm   = wave >> 1;            // 0..3 : M sub-tile
  const int wn   = wave & 1;             // 0..1 : N sub-tile
  const int lrow = lane & 15;
  const int khi  = (lane >> 4) << 1;     // lanes 16-31 hold K+2, K+3
  const int blockM = blockIdx.y * 64;
  const int blockN = blockIdx.x * 128;

  v8f acc[4];
#pragma unroll
  for (int t = 0; t < 4; ++t)
#pragma unroll
    for (int i = 0; i < 8; ++i) acc[t][i] = 0.0f;

  for (int k0 = 0; k0 < DCH; k0 += 32) {
    // Stage A tile: 64x32 floats = 512 float4 chunks, 2 per thread
#pragma unroll
    for (int i = 0; i < 2; ++i) {
      int off = (tid + i * 256) * 4;
      int r = off >> 5, c = off & 31;
      int gr = blockM + r;
      fvec4 v;
      v[0] = v[1] = v[2] = v[3] = 0.0f;
      if (gr < Nrows) v = *(const fvec4*)(X + (long)gr * DCH + k0 + c);
      *(fvec4*)(As + off) = v;
    }
    // Stage B tile transposed: read W row-major (float4 along n), write [n][k]
#pragma unroll
    for (int i = 0; i < 4; ++i) {
      int off = (tid + i * 256) * 4;
      int kr = off >> 7, c = off & 127;
      fvec4 v = *(const fvec4*)(W + (long)(k0 + kr) * DCH + blockN + c);
#pragma unroll
      for (int j = 0; j < 4; ++j) Bs[(c + j) * BS_STRIDE + kr] = v[j];
    }
    __syncthreads();

#pragma unroll
    for (int kk = 0; kk < 32; kk += 4) {
      // A fragment (16x4): contiguous pair -> single ds_load_b64
      v2f a = *(const v2f*)(As + (wm * 16 + lrow) * 32 + kk + khi);
#pragma unroll
      for (int t = 0; t < 4; ++t) {
        int nc = wn * 64 + t * 16 + lrow;
        // B fragment (4x16): contiguous pair in transposed tile -> ds_load_b64
        v2f b = *(const v2f*)(Bs + nc * BS_STRIDE + kk + khi);
        acc[t] = __builtin_amdgcn_wmma_f32_16x16x4_f32(
            /*neg_a=*/false, a, /*neg_b=*/false, b,
            /*c_mod=*/(short)0, acc[t], /*reuse_a=*/false, /*reuse_b=*/false);
      }
    }
    __syncthreads();
  }

  // Epilogue: D layout VGPR i -> rows (i, i+8); lanes 16-31 are the +8 rows.
  const int rowbase = blockM + wm * 16 + ((lane >> 4) << 3);
#pragma unroll
  for (int i = 0; i < 8; ++i) {
    int row = rowbase + i;
    if (row < Nrows) {
#pragma unroll
      for (int t = 0; t < 4; ++t) {
        int col = blockN + wn * 64 + t * 16 + lrow;
        Y[(long)row * DCH + col] = acc[t][i] + bias[col];
      }
    }
  }
}

// ---------------------------------------------------------------------------
// BN column statistics: each block accumulates 128 rows, thread t owns cols
// t and t+256 (coalesced row reads), one atomicAdd per block per column.
// ---------------------------------------------------------------------------
__global__ __launch_bounds__(256) void bn_stats(
    const float* __restrict__ Y, int Nrows,
    float* __restrict__ colsum, float* __restrict__ colsumsq) {
  int t = threadIdx.x;
  int r0 = blockIdx.x * 128;
  int r1 = r0 + 128 < Nrows ? r0 + 128 : Nrows;
  float s0 = 0.f, s1 = 0.f, q0 = 0.f, q1 = 0.f;
  for (int r = r0; r < r1; ++r) {
    float a = Y[(long)r * DCH + t];
    float b = Y[(long)r * DCH + t + 256];
    s0 += a; q0 += a * a;
    s1 += b; q1 += b * b;
  }
  atomicAdd(colsum + t, s0);
  atomicAdd(colsum + t + 256, s1);
  atomicAdd(colsumsq + t, q0);
  atomicAdd(colsumsq + t + 256, q1);
}

__global__ void bn_coef(const float* __restrict__ colsum,
                        const float* __restrict__ colsumsq,
                        const float* __restrict__ gamma,
                        const float* __restrict__ beta,
                        float* __restrict__ scale, float* __restrict__ shift,
                        int Nrows) {
  int c = threadIdx.x;
  float inv = 1.0f / (float)Nrows;
  float mu  = colsum[c] * inv;
  float var = colsumsq[c] * inv - mu * mu;
  float s = rsqrtf(var + BN_EPS) * gamma[c];
  scale[c] = s;
  shift[c] = beta[c] - mu * s;
}

__global__ __launch_bounds__(256) void bn_apply(
    float* __restrict__ Y, const float* __restrict__ scale,
    const float* __restrict__ shift, long total4) {
  long i = (long)blockIdx.x * 256 + threadIdx.x;
  long stride = (long)gridDim.x * 256;
  for (; i < total4; i += stride) {
    int c0 = (int)((i * 4) & (DCH - 1));
    fvec4 y = *(fvec4*)(Y + i * 4);
    fvec4 s = *(const fvec4*)(scale + c0);
    fvec4 h = *(const fvec4*)(shift + c0);
    y = y * s + h;
    *(fvec4*)(Y + i * 4) = y;
  }
}

// ---------------------------------------------------------------------------
// Degrees: segment softmax of all-ones == 1/count, so only counts are needed.
// ---------------------------------------------------------------------------
__global__ __launch_bounds__(256) void degrees(
    const int* __restrict__ v_idx, const int* __restrict__ e_idx,
    float* __restrict__ deg_v, float* __restrict__ deg_e, int M) {
  int i = blockIdx.x * 256 + threadIdx.x;
  if (i < M) {
    atomicAdd(deg_v + v_idx[i], 1.0f);
    atomicAdd(deg_e + e_idx[i], 1.0f);
  }
}

// ---------------------------------------------------------------------------
// Weighted scatter-add: dst[dst_idx[m]] += src[src_idx[m]] / deg[dst_idx[m]]
// One block per incidence, 128 threads x 4 columns. Atomics resolve in L2
// (both HE (20 MB) and X (102 MB) fit in the 192 MB L2).
// ---------------------------------------------------------------------------
__global__ __launch_bounds__(128) void scatter_wsum(
    const float* __restrict__ src, const int* __restrict__ src_idx,
    const int* __restrict__ dst_idx, const float* __restrict__ deg,
    float* __restrict__ dst, int M) {
  int m = blockIdx.x;
  if (m >= M) return;
  int t = threadIdx.x;
  int s = src_idx[m];
  int d = dst_idx[m];
  float w = 1.0f / deg[d];
  const float* sp = src + (long)s * DCH;
  float* dp = dst + (long)d * DCH;
#pragma unroll
  for (int j = 0; j < 4; ++j) {
    int c = t + j * 128;
    atomicAdd(dp + c, w * sp[c]);
  }
}

extern "C" void kernel_launch(void* const* d_in, const int* in_sizes, int n_in,
                              void* d_out, int out_size, void* d_ws, size_t ws_size,
                              hipStream_t stream) {
  const float* X     = (const float*)d_in[0];
  const float* W     = (const float*)d_in[1];
  const float* bias  = (const float*)d_in[2];
  const float* gamma = (const float*)d_in[3];
  const float* beta  = (const float*)d_in[4];
  const int*   v_idx = (const int*)d_in[5];
  const int*   e_idx = (const int*)d_in[6];

  const int Nrows = in_sizes[0] / DCH;   // 50000
  const int M     = in_sizes[5];         // 200000
  const int E     = 10000;               // hyperedges (reference constant)

  float* out = (float*)d_out;
  float* Y   = out;                           // reuse X_out region as GEMM scratch
  float* HE  = out + (size_t)Nrows * DCH;     // second tuple output

  float* ws       = (float*)d_ws;
  float* colsum   = ws;            // 512
  float* colsumsq = ws + 512;      // 512
  float* scale    = ws + 1024;     // 512
  float* shift    = ws + 1536;     // 512
  float* deg_e    = ws + 2048;     // E
  float* deg_v    = deg_e + E;     // Nrows

  // Zero accumulation targets (stream-ordered, graph-capturable).
  hipMemsetAsync(ws, 0, (size_t)(2048 + E + Nrows) * sizeof(float), stream);
  hipMemsetAsync(HE, 0, (size_t)E * DCH * sizeof(float), stream);

  // 1) Y = X @ W + b  (WMMA fp32)
  dim3 ggrid(DCH / 128, (Nrows + 63) / 64);
  gemm_wmma_f32<<<ggrid, 256, 0, stream>>>(X, W, bias, Y, Nrows);

  // 2) BatchNorm (training-mode batch stats, biased variance)
  bn_stats<<<(Nrows + 127) / 128, 256, 0, stream>>>(Y, Nrows, colsum, colsumsq);
  bn_coef<<<1, DCH, 0, stream>>>(colsum, colsumsq, gamma, beta, scale, shift, Nrows);
  long total4 = (long)Nrows * DCH / 4;
  bn_apply<<<2048, 256, 0, stream>>>(Y, scale, shift, total4);

  // 3) Segment counts (softmax of ones == 1/count)
  degrees<<<(M + 255) / 256, 256, 0, stream>>>(v_idx, e_idx, deg_v, deg_e, M);

  // 4) v2e: HE[e] += Y[v] / deg_e[e]
  scatter_wsum<<<M, 128, 0, stream>>>(Y, v_idx, e_idx, deg_e, HE, M);

  // 5) X_out[v] += HE[e] / deg_v[v]  (Y is dead now; zero its region first)
  hipMemsetAsync(Y, 0, (size_t)Nrows * DCH * sizeof(float), stream);
  scatter_wsum<<<M, 128, 0, stream>>>(HE, e_idx, v_idx, deg_v, out, M);
}